// SSIM_4552665334004
// MI455X (gfx1250) — compile-verified
//
#include <hip/hip_runtime.h>
#include <stdint.h>

typedef unsigned int u32x4 __attribute__((ext_vector_type(4)));
typedef int          i32x8 __attribute__((ext_vector_type(8)));
typedef int          i32x4 __attribute__((ext_vector_type(4)));

#define H        512
#define W        512
#define OH       510
#define OW       510
#define PLANES   96          // 32 * 3
#define TILE_IN  64
#define TILE_OUT 62
#define LDS_PITCH 65         // 64 + 1 pad DWORD inserted by TDM pad feature
#define TILES_X  9           // ceil(510/62)
#define TILES_Y  9
#define BAND     16          // rows of outputs per thread-band (4 bands x 64 cols = 256 thr)

// Build a 2D TDM descriptor and issue tensor_load_to_lds:
//   tile: TILE_IN x TILE_IN fp32, memory row stride = W floats,
//   LDS rows padded to LDS_PITCH floats (pad_interval=64 DW, pad=1 DW).
__device__ __forceinline__ void tdm_load_tile(const float* gbase, unsigned lds_off,
                                              int tx, int ty) {
  uint64_t gaddr = (uint64_t)(uintptr_t)(gbase + (size_t)ty * W + tx);

  u32x4 g0;
  g0[0] = 1u;                                   // count=1, is_restore=0, gather=0
  g0[1] = lds_off;                              // LDS byte address
  g0[2] = (unsigned)(gaddr & 0xFFFFFFFFull);    // global_addr[31:0]
  g0[3] = (unsigned)(gaddr >> 32) | (2u << 30); // global_addr[56:32], type=2 (image)

  // Remaining extent from the tile origin so TDM OOB logic zero-fills the halo
  // that spills past the 512x512 plane on edge tiles.
  int td0 = W - tx;   // dim0 (contiguous x) extent
  int td1 = H - ty;   // dim1 (rows) extent

  i32x8 g1;
  g1[0] = (2 << 16)      // data_size = 4 bytes
        | (1 << 20)      // pad_enable
        | (5 << 22);     // pad_interval: pad after every 64 DWORDs
                         // pad_amount bits[31:25] = 0 -> 1 DWORD pad
  g1[1] = (td0 & 0xFFFF) << 16;                                   // tensor_dim0[15:0]
  g1[2] = ((td0 >> 16) & 0xFFFF) | ((td1 & 0xFFFF) << 16);        // dim0 hi | dim1 lo
  g1[3] = ((td1 >> 16) & 0xFFFF) | (TILE_IN << 16);               // dim1 hi | tile_dim0
  g1[4] = TILE_IN;                                                // tile_dim1 (tile_dim2=0)
  g1[5] = W;                                                      // tensor_dim0_stride lo
  g1[6] = 0;
  g1[7] = 0;

  i32x4 z4 = {0, 0, 0, 0};                      // groups 2/3 unused (2D tensor)
  i32x8 z8 = {0, 0, 0, 0, 0, 0, 0, 0};
  __builtin_amdgcn_tensor_load_to_lds(g0, g1, z4, z4, z8, 0);
}

// Horizontal 3-tap sums of row r at column ox for both images (6 LDS reads).
__device__ __forceinline__ void hsum_row(const float* __restrict__ tX,
                                         const float* __restrict__ tY,
                                         int r, int ox,
                                         float& ha, float& hb, float& haa,
                                         float& hbb, float& hab) {
  const int base = r * LDS_PITCH + ox;
  const float a0 = tX[base],     a1 = tX[base + 1], a2 = tX[base + 2];
  const float b0 = tY[base],     b1 = tY[base + 1], b2 = tY[base + 2];
  ha  = a0 + a1 + a2;
  hb  = b0 + b1 + b2;
  haa = a0 * a0 + a1 * a1 + a2 * a2;
  hbb = b0 * b0 + b1 * b1 + b2 * b2;
  hab = a0 * b0 + a1 * b1 + a2 * b2;
}

__global__ void __launch_bounds__(256)
ssim_tile_kernel(const float* __restrict__ x, const float* __restrict__ y,
                 float* __restrict__ partial) {
  __shared__ float tX[TILE_IN * LDS_PITCH];
  __shared__ float tY[TILE_IN * LDS_PITCH];
  __shared__ float red[256];

  const int tile  = blockIdx.x;
  const int plane = blockIdx.y;
  const int tx = (tile % TILES_X) * TILE_OUT;
  const int ty = (tile / TILES_X) * TILE_OUT;
  const size_t plane_off = (size_t)plane * (size_t)H * (size_t)W;

  // One wave issues both async tensor DMAs and waits on TENSORcnt;
  // the workgroup barrier then publishes the LDS tiles to all 8 waves.
  if (threadIdx.x < 32) {
    tdm_load_tile(x + plane_off, (unsigned)(uintptr_t)(void*)tX, tx, ty);
    tdm_load_tile(y + plane_off, (unsigned)(uintptr_t)(void*)tY, tx, ty);
    __builtin_amdgcn_s_wait_tensorcnt(0);
  }
  __syncthreads();

  const float inv9 = 1.0f / 9.0f;
  const float C1 = 1e-4f;   // 0.01^2
  const float C2 = 9e-4f;   // 0.03^2

  // 2D thread map: 64 columns x 4 row-bands. No per-pixel div/guards:
  // loop bounds are clamped against the 510x510 output up front.
  const int ox   = (int)(threadIdx.x & 63);
  const int band = (int)(threadIdx.x >> 6);
  const int row0 = band * BAND;
  const int rows_in_tile = min(TILE_OUT, OH - ty);      // valid output rows here
  const int row1 = min(row0 + BAND, rows_in_tile);
  const bool col_ok = (ox < TILE_OUT) && (tx + ox < OW);

  float acc = 0.0f;
  if (col_ok && row0 < row1) {
    // Ring of per-row horizontal sums for the 5 statistics.
    float ha0, hb0, haa0, hbb0, hab0;
    float ha1, hb1, haa1, hbb1, hab1;
    float ha2, hb2, haa2, hbb2, hab2;
    hsum_row(tX, tY, row0,     ox, ha0, hb0, haa0, hbb0, hab0);
    hsum_row(tX, tY, row0 + 1, ox, ha1, hb1, haa1, hbb1, hab1);

    for (int oy = row0; oy < row1; ++oy) {
      hsum_row(tX, tY, oy + 2, ox, ha2, hb2, haa2, hbb2, hab2);

      const float sx  = ha0  + ha1  + ha2;
      const float sy  = hb0  + hb1  + hb2;
      const float sxx = haa0 + haa1 + haa2;
      const float syy = hbb0 + hbb1 + hbb2;
      const float sxy = hab0 + hab1 + hab2;

      const float mux = sx * inv9;
      const float muy = sy * inv9;
      const float sigx  = sxx * inv9 - mux * mux;
      const float sigy  = syy * inv9 - muy * muy;
      const float sigxy = sxy * inv9 - mux * muy;
      const float num = (2.0f * mux * muy + C1) * (2.0f * sigxy + C2);
      const float den = (mux * mux + muy * muy + C1) * (sigx + sigy + C2);
      acc += 1.0f - num / den;

      // rotate ring
      ha0 = ha1;  hb0 = hb1;  haa0 = haa1;  hbb0 = hbb1;  hab0 = hab1;
      ha1 = ha2;  hb1 = hb2;  haa1 = haa2;  hbb1 = hbb2;  hab1 = hab2;
    }
  }

  // Deterministic block reduction.
  red[threadIdx.x] = acc;
  __syncthreads();
#pragma unroll
  for (int off = 128; off > 0; off >>= 1) {
    if (threadIdx.x < off) red[threadIdx.x] += red[threadIdx.x + off];
    __syncthreads();
  }
  if (threadIdx.x == 0)
    partial[(size_t)blockIdx.y * gridDim.x + blockIdx.x] = red[0];
}

__global__ void __launch_bounds__(256)
ssim_reduce_kernel(const float* __restrict__ partial, float* __restrict__ out,
                   int n, float scale) {
  __shared__ float red[256];
  float s = 0.0f;
  for (int i = threadIdx.x; i < n; i += 256) s += partial[i];
  red[threadIdx.x] = s;
  __syncthreads();
#pragma unroll
  for (int off = 128; off > 0; off >>= 1) {
    if (threadIdx.x < off) red[threadIdx.x] += red[threadIdx.x + off];
    __syncthreads();
  }
  if (threadIdx.x == 0) out[0] = red[0] * scale;
}

extern "C" void kernel_launch(void* const* d_in, const int* in_sizes, int n_in,
                              void* d_out, int out_size, void* d_ws, size_t ws_size,
                              hipStream_t stream) {
  (void)in_sizes; (void)n_in; (void)out_size; (void)ws_size;
  const float* x = (const float*)d_in[0];
  const float* y = (const float*)d_in[1];
  float* out     = (float*)d_out;
  float* partial = (float*)d_ws;   // TILES_X*TILES_Y*PLANES = 7776 floats

  dim3 grid(TILES_X * TILES_Y, PLANES);
  ssim_tile_kernel<<<grid, 256, 0, stream>>>(x, y, partial);

  const int   n     = TILES_X * TILES_Y * PLANES;
  const float scale = 1.0f / ((float)PLANES * (float)OH * (float)OW);
  ssim_reduce_kernel<<<1, 256, 0, stream>>>(partial, out, n, scale);
}